// Generator2DLUT_identity_32693291057269
// MI455X (gfx1250) — compile-verified
//
#include <hip/hip_runtime.h>

typedef float v4f __attribute__((ext_vector_type(4)));

#define LUT_D   251
#define LUT_DM1 250.0f
#define IMG_HW  (512 * 512)
#define N_PIX   (32 * IMG_HW)     // 8,388,608 pixels
#define N_CHUNK (N_PIX / 4)       // 2,097,152 four-pixel chunks
#define MAIN_BLOCKS  4096
#define MAIN_THREADS 256
// grid-stride in pixels = MAIN_BLOCKS*MAIN_THREADS*4 = 16 full images
// => next chunk's coordinate base is exactly base + 16*(2*HW) = base + 32*HW.
#define NEXT_OFF ((size_t)32 * IMG_HW)

// ---------------------------------------------------------------------------
// Main kernel (placed first so the disasm snippet shows it): 4 pixels/thread.
// Streaming planes: non-temporal B128 loads/stores (single-use data — keep
// WGP$/L2 for the gather table). Gathers: 2 aligned B128 loads per pixel from
// the 16B-packed LUT (each load = exactly the 4 floats one corner-row needs).
// global_prefetch_b8 covers the next grid-stride chunk of coordinates.
// ---------------------------------------------------------------------------
__global__ void lut2d_bilinear_kernel(const float* __restrict__ x,
                                      const v4f* __restrict__ tbl,
                                      float* __restrict__ out) {
    const int stride = gridDim.x * blockDim.x;
    for (int t = blockIdx.x * blockDim.x + threadIdx.x; t < N_CHUNK; t += stride) {
        int pix = t * 4;
        int b   = pix / IMG_HW;                       // power-of-two: shift
        int r   = pix - b * IMG_HW;
        size_t base = (size_t)(b * 2) * IMG_HW + r;   // plane a; plane b at +HW

        // Prefetch next grid-stride chunk's coordinates (global_prefetch_b8).
        if (t + stride < N_CHUNK) {
            __builtin_prefetch(x + base + NEXT_OFF, 0, 0);
            __builtin_prefetch(x + base + NEXT_OFF + IMG_HW, 0, 0);
        }

        v4f a4 = __builtin_nontemporal_load((const v4f*)(x + base));
        v4f b4 = __builtin_nontemporal_load((const v4f*)(x + base + IMG_HW));

        v4f o0, o1;
#pragma unroll
        for (int p = 0; p < 4; ++p) {
            float a  = fminf(fmaxf(a4[p], 0.0f), 1.0f) * LUT_DM1;
            float bb = fminf(fmaxf(b4[p], 0.0f), 1.0f) * LUT_DM1;
            int i0 = min((int)a,  LUT_D - 2);   // a,bb >= 0: trunc == floor
            int j0 = min((int)bb, LUT_D - 2);
            float fa = a  - (float)i0;
            float fb = bb - (float)j0;
            int k = i0 * LUT_D + j0;

            v4f r0 = tbl[k];           // {v00c0, v00c1, v01c0, v01c1}
            v4f r1 = tbl[k + LUT_D];   // {v10c0, v10c1, v11c0, v11c1}

            float wa0 = 1.0f - fa, wb0 = 1.0f - fb;
            float w00 = wa0 * wb0, w01 = wa0 * fb;
            float w10 = fa * wb0,  w11 = fa * fb;

            o0[p] = r0.x * w00 + r0.z * w01 + r1.x * w10 + r1.z * w11;
            o1[p] = r0.y * w00 + r0.w * w01 + r1.y * w10 + r1.w * w11;
        }

        __builtin_nontemporal_store(o0, (v4f*)(out + base));
        __builtin_nontemporal_store(o1, (v4f*)(out + base + IMG_HW));
    }
}

// ---------------------------------------------------------------------------
// Pack LUT (2,251,251) into 16B entries: tbl[i*251+j] = {c0(i,j), c1(i,j),
// c0(i,j+1), c1(i,j+1)}. One B128 gather then yields all 4 floats a bilinear
// corner-row needs (both channels, both adjacent columns) with zero waste.
// ---------------------------------------------------------------------------
__global__ void pack_lut_kernel(const float* __restrict__ lut,
                                v4f* __restrict__ tbl) {
    int k = blockIdx.x * blockDim.x + threadIdx.x;
    if (k >= LUT_D * LUT_D) return;
    int i = k / LUT_D;
    int j = k - i * LUT_D;
    int jn = (j < LUT_D - 1) ? j + 1 : j;      // last column entry is never read
    const float* __restrict__ c0 = lut;                  // channel 0 plane
    const float* __restrict__ c1 = lut + LUT_D * LUT_D;  // channel 1 plane
    v4f p;
    p.x = c0[i * LUT_D + j];
    p.y = c1[i * LUT_D + j];
    p.z = c0[i * LUT_D + jn];
    p.w = c1[i * LUT_D + jn];
    tbl[k] = p;
}

// ---------------------------------------------------------------------------
// Fallback (only if workspace < 1 MB): direct scalar gathers from original LUT.
// ---------------------------------------------------------------------------
__global__ void lut2d_direct_kernel(const float* __restrict__ x,
                                    const float* __restrict__ lut,
                                    float* __restrict__ out) {
    int t = blockIdx.x * blockDim.x + threadIdx.x;
    if (t >= N_CHUNK) return;
    int pix = t * 4;
    int b   = pix / IMG_HW;
    int r   = pix - b * IMG_HW;
    size_t base = (size_t)(b * 2) * IMG_HW + r;

    v4f a4 = __builtin_nontemporal_load((const v4f*)(x + base));
    v4f b4 = __builtin_nontemporal_load((const v4f*)(x + base + IMG_HW));
    const float* __restrict__ c0 = lut;
    const float* __restrict__ c1 = lut + LUT_D * LUT_D;

    v4f o0, o1;
#pragma unroll
    for (int p = 0; p < 4; ++p) {
        float a  = fminf(fmaxf(a4[p], 0.0f), 1.0f) * LUT_DM1;
        float bb = fminf(fmaxf(b4[p], 0.0f), 1.0f) * LUT_DM1;
        int i0 = min((int)a,  LUT_D - 2);
        int j0 = min((int)bb, LUT_D - 2);
        float fa = a  - (float)i0;
        float fb = bb - (float)j0;
        int k = i0 * LUT_D + j0;

        float wa0 = 1.0f - fa, wb0 = 1.0f - fb;
        float w00 = wa0 * wb0, w01 = wa0 * fb;
        float w10 = fa * wb0,  w11 = fa * fb;

        o0[p] = c0[k] * w00 + c0[k + 1] * w01 + c0[k + LUT_D] * w10 + c0[k + LUT_D + 1] * w11;
        o1[p] = c1[k] * w00 + c1[k + 1] * w01 + c1[k + LUT_D] * w10 + c1[k + LUT_D + 1] * w11;
    }
    __builtin_nontemporal_store(o0, (v4f*)(out + base));
    __builtin_nontemporal_store(o1, (v4f*)(out + base + IMG_HW));
}

extern "C" void kernel_launch(void* const* d_in, const int* in_sizes, int n_in,
                              void* d_out, int out_size, void* d_ws, size_t ws_size,
                              hipStream_t stream) {
    const float* x   = (const float*)d_in[0];   // (32, 2, 512, 512) fp32
    const float* lut = (const float*)d_in[1];   // (1, 2, 251, 251)  fp32
    float* out = (float*)d_out;                 // (32, 2, 512, 512) fp32

    const size_t tbl_bytes = (size_t)LUT_D * LUT_D * sizeof(v4f);  // ~1.0 MB

    if (ws_size >= tbl_bytes) {
        v4f* tbl = (v4f*)d_ws;
        pack_lut_kernel<<<(LUT_D * LUT_D + 255) / 256, 256, 0, stream>>>(lut, tbl);
        // 4096 blocks x 8 wave32s: each thread runs exactly 2 grid-stride
        // iterations, so the prefetch path always has a live next chunk on
        // iteration 1; 32K waves is plenty to saturate HBM.
        lut2d_bilinear_kernel<<<MAIN_BLOCKS, MAIN_THREADS, 0, stream>>>(x, tbl, out);
    } else {
        lut2d_direct_kernel<<<(N_CHUNK + 255) / 256, 256, 0, stream>>>(x, lut, out);
    }
}